// GIN_6554120094213
// MI455X (gfx1250) — compile-verified
//
#include <hip/hip_runtime.h>

#define DIM 128
#define NG 128
#define NLAYERS 3
#define BN_EPS 1e-5f

// LDS row strides (dwords) produced by TDM pad feature
#define RSZ 130   // Z/Y tile: 128 + 2  (8B aligned rows, conflict-free strided reads)
#define RSW 136   // W tiles:  128 + 8  (lane-half B-fragment reads hit disjoint banks)
// LDS layout (dwords): sZ/sY[128*130] | sW[128*136] | sSum[128] | sSq[128]
#define LDS_W_OFF   (128 * RSZ)              // 16640
#define LDS_SUM_OFF (LDS_W_OFF + 128 * RSW)  // 34048
#define LDS_SQ_OFF  (LDS_SUM_OFF + DIM)      // 34176
#define LDS_DWORDS  (LDS_SQ_OFF + DIM)       // 34304
#define LDS_BYTES   (LDS_DWORDS * 4)         // 137216 (<160KB -> 2 blocks/WGP)

typedef float v2f __attribute__((ext_vector_type(2)));
typedef float v8f __attribute__((ext_vector_type(8)));
typedef unsigned int u32x4 __attribute__((ext_vector_type(4)));
typedef int i32x4 __attribute__((ext_vector_type(4)));
typedef int i32x8 __attribute__((ext_vector_type(8)));

static __device__ __forceinline__ unsigned ldsOff(const void* p) {
    return (unsigned)(size_t)p;   // low 32 bits of LDS aperture addr = byte offset
}

// Tensor DMA: 2D tile (data_size = 4B) global -> LDS, optional per-row LDS padding.
// D# encoding per CDNA5 ISA 8.3/8.4: group0 = {count|flags, lds_addr, global_addr, type=2},
// group1 = {cfg, tensor_dim0/1, tile_dim0/1, tensor_dim0_stride}.
static __device__ __forceinline__ void tdm_load_2d(
    unsigned ldsAddr, const void* gptr,
    unsigned td0, unsigned td1, unsigned tl0, unsigned tl1,
    unsigned stride0, unsigned padAmtCode, bool padEn) {
    unsigned long long ga = (unsigned long long)gptr;
    u32x4 g0 = { 1u,                                     // count=1, is_restore=0, gather=0
                 ldsAddr,                                // [63:32] lds byte addr
                 (unsigned)ga,                           // [95:64] global addr lo
                 ((unsigned)(ga >> 32) & 0x01FFFFFFu) | 0x80000000u };  // addr hi | type=2
    unsigned w0 = (2u << 16);                            // data_size = 4 bytes
    if (padEn) w0 |= (1u << 20) | (6u << 22) | (padAmtCode << 25);  // pad every 128 dw
    i32x8 g1 = { (int)w0,
                 (int)((td0 & 0xFFFFu) << 16),                            // dim0 lo
                 (int)(((td0 >> 16) & 0xFFFFu) | ((td1 & 0xFFFFu) << 16)),// dim0 hi|dim1 lo
                 (int)(((td1 >> 16) & 0xFFFFu) | ((tl0 & 0xFFFFu) << 16)),// dim1 hi|tile0
                 (int)(tl1 & 0xFFFFu),                                    // tile1 (tile2=0)
                 (int)stride0, 0, 0 };                                    // dim0 stride
    i32x4 gz = {0, 0, 0, 0};
#if defined(__clang_major__) && (__clang_major__ >= 23)
    i32x8 gz8 = {0, 0, 0, 0, 0, 0, 0, 0};
    __builtin_amdgcn_tensor_load_to_lds(g0, g1, gz, gz, gz8, 0);
#else
    __builtin_amdgcn_tensor_load_to_lds(g0, g1, gz, gz, 0);
#endif
}

// ---------------------------------------------------------------- utilities
__global__ __launch_bounds__(256) void k_zero(float* __restrict__ p, int n) {
    int i = blockIdx.x * 256 + threadIdx.x;
    if (i < n) p[i] = 0.f;
}

__global__ __launch_bounds__(256) void k_copy4(const float4* __restrict__ s,
                                               float4* __restrict__ d, int n4) {
    int i = blockIdx.x * 256 + threadIdx.x;
    if (i < n4) d[i] = s[i];
}

// --------------------------------------------------- edge scatter (segment_sum)
__global__ __launch_bounds__(256) void k_scatter(const float* __restrict__ h,
                                                 const int* __restrict__ edge,
                                                 float* __restrict__ z, int nEdges) {
    int t = blockIdx.x * 256 + threadIdx.x;
    int e = t >> 5;
    if (e >= nEdges) return;
    int c = (t & 31) << 2;
    int s = edge[e];
    int d = edge[nEdges + e];
    const float4 v = *(const float4*)(h + (size_t)s * DIM + c);
    float* zp = z + (size_t)d * DIM + c;
    unsafeAtomicAdd(zp + 0, v.x);
    unsafeAtomicAdd(zp + 1, v.y);
    unsafeAtomicAdd(zp + 2, v.z);
    unsafeAtomicAdd(zp + 3, v.w);
}

// ------------------------------------------- fused MLP: relu(relu(Z@W1+b1)@W2+b2)
// 8 waves/block, 16 rows/wave, TDM tile loads, fp32 WMMA 16x16x4, fused BN stats.
__global__ __launch_bounds__(256) void k_gin_mlp(
    const float* __restrict__ Z, const float* __restrict__ W1,
    const float* __restrict__ b1, const float* __restrict__ W2,
    const float* __restrict__ b2, float* __restrict__ Out,
    float* __restrict__ accS, float* __restrict__ accQ, int nNodes) {
    extern __shared__ float smem[];
    float* sZ   = smem;                 // 128 x RSZ : Z tile, reused as Y1 tile
    float* sW   = smem + LDS_W_OFF;     // 128 x RSW : W1, then W2
    float* sSum = smem + LDS_SUM_OFF;
    float* sSq  = smem + LDS_SQ_OFF;

    const int tid  = threadIdx.x;
    const int wave = tid >> 5;
    const int lane = tid & 31;
    const int half = lane >> 4;         // 0: lanes 0-15, 1: lanes 16-31
    const int l16  = lane & 15;
    const int rowBase0 = blockIdx.x * 128;
    const int rowBase  = rowBase0 + wave * 16;

    if (tid < DIM) { sSum[tid] = 0.f; sSq[tid] = 0.f; }

    if (wave == 0) {
        // Z tile: tensor_dim1 = remaining rows -> hardware OOB-zero for tail block
        tdm_load_2d(ldsOff(sZ), Z + (size_t)rowBase0 * DIM,
                    DIM, (unsigned)(nNodes - rowBase0), DIM, 128, DIM, 1u, true);
        tdm_load_2d(ldsOff(sW), W1, DIM, DIM, DIM, DIM, DIM, 7u, true);
        __builtin_amdgcn_s_wait_tensorcnt(0);
    }
    __syncthreads();

    // A-fragments: lane holds row (rowBase+l16); K pair selected by lane half.
    v2f a[32];
    {
        const float* zr = sZ + (size_t)(wave * 16 + l16) * RSZ;
        #pragma unroll
        for (int ks = 0; ks < 32; ++ks)
            a[ks] = *(const v2f*)(zr + ks * 4 + half * 2);
    }

    // ---- GEMM1: Y1 = relu(Z @ W1 + b1) -> back into sZ region (own rows only)
    #pragma unroll
    for (int ot = 0; ot < 8; ++ot) {
        const int ob = ot * 16;
        v8f acc = {0.f, 0.f, 0.f, 0.f, 0.f, 0.f, 0.f, 0.f};
        #pragma unroll
        for (int ks = 0; ks < 32; ++ks) {
            int k = ks * 4 + half * 2;
            v2f bf;
            bf.x = sW[k * RSW + ob + l16];
            bf.y = sW[(k + 1) * RSW + ob + l16];
            acc = __builtin_amdgcn_wmma_f32_16x16x4_f32(
                false, a[ks], false, bf, (short)0, acc, false, false);
        }
        float bias = b1[ob + l16];
        #pragma unroll
        for (int v = 0; v < 8; ++v) {
            float y = acc[v] + bias;
            y = y > 0.f ? y : 0.f;
            sZ[(wave * 16 + v + half * 8) * RSZ + ob + l16] = y;
        }
    }
    __syncthreads();                    // all waves done with W1 (and Y1 complete)

    if (wave == 0) {                    // overwrite W1 region with W2
        tdm_load_2d(ldsOff(sW), W2, DIM, DIM, DIM, DIM, DIM, 7u, true);
        __builtin_amdgcn_s_wait_tensorcnt(0);
    }
    __syncthreads();

    // reload A-fragments from Y1 (layout conversion D -> A via LDS)
    {
        const float* yr = sZ + (size_t)(wave * 16 + l16) * RSZ;
        #pragma unroll
        for (int ks = 0; ks < 32; ++ks)
            a[ks] = *(const v2f*)(yr + ks * 4 + half * 2);
    }

    // ---- GEMM2: Y2 = relu(Y1 @ W2 + b2) -> global, + BN partial sums
    #pragma unroll
    for (int ot = 0; ot < 8; ++ot) {
        const int ob = ot * 16;
        v8f acc = {0.f, 0.f, 0.f, 0.f, 0.f, 0.f, 0.f, 0.f};
        #pragma unroll
        for (int ks = 0; ks < 32; ++ks) {
            int k = ks * 4 + half * 2;
            v2f bf;
            bf.x = sW[k * RSW + ob + l16];
            bf.y = sW[(k + 1) * RSW + ob + l16];
            acc = __builtin_amdgcn_wmma_f32_16x16x4_f32(
                false, a[ks], false, bf, (short)0, acc, false, false);
        }
        float bias = b2[ob + l16];
        float s = 0.f, q = 0.f;
        #pragma unroll
        for (int v = 0; v < 8; ++v) {
            int grow = rowBase + v + half * 8;
            float y = acc[v] + bias;
            y = y > 0.f ? y : 0.f;
            if (grow < nNodes) {
                Out[(size_t)grow * DIM + ob + l16] = y;
                s += y;
                q += y * y;
            }
        }
        atomicAdd(&sSum[ob + l16], s);  // ds_add_f32
        atomicAdd(&sSq[ob + l16], q);
    }
    __syncthreads();
    if (tid < DIM) {
        unsafeAtomicAdd(&accS[tid], sSum[tid]);
        unsafeAtomicAdd(&accQ[tid], sSq[tid]);
    }
}

// ---------------------------------------------------------------- BN finalize
__global__ __launch_bounds__(DIM) void k_bn_finalize(
    const float* __restrict__ accS, const float* __restrict__ accQ,
    const float* __restrict__ gamma, const float* __restrict__ beta,
    float* __restrict__ scale, float* __restrict__ shift, int nNodes) {
    int f = threadIdx.x;
    float inv_n = 1.0f / (float)nNodes;
    float mu  = accS[f] * inv_n;
    float var = accQ[f] * inv_n - mu * mu;
    float sc  = gamma[f] * rsqrtf(var + BN_EPS);
    scale[f] = sc;
    shift[f] = beta[f] - mu * sc;
}

// ------------------------------------------------ BN apply + global_add_pool
__global__ __launch_bounds__(256) void k_bn_pool(
    const float* __restrict__ z, const float* __restrict__ scale,
    const float* __restrict__ shift, const int* __restrict__ batch,
    float* __restrict__ hOut, float* __restrict__ pool, int layer, int nNodes) {
    int t = blockIdx.x * 256 + threadIdx.x;
    int node = t >> 5;
    if (node >= nNodes) return;
    int c = (t & 31) << 2;
    const float4 v  = *(const float4*)(z + (size_t)node * DIM + c);
    const float4 sc = *(const float4*)(scale + c);
    const float4 sh = *(const float4*)(shift + c);
    float4 h;
    h.x = v.x * sc.x + sh.x;
    h.y = v.y * sc.y + sh.y;
    h.z = v.z * sc.z + sh.z;
    h.w = v.w * sc.w + sh.w;
    *(float4*)(hOut + (size_t)node * DIM + c) = h;
    int g = batch[node];
    float* pp = pool + (size_t)g * (NLAYERS * DIM) + layer * DIM + c;
    unsafeAtomicAdd(pp + 0, h.x);
    unsafeAtomicAdd(pp + 1, h.y);
    unsafeAtomicAdd(pp + 2, h.z);
    unsafeAtomicAdd(pp + 3, h.w);
}

// ---------------------------------------------------------------- launcher
extern "C" void kernel_launch(void* const* d_in, const int* in_sizes, int n_in,
                              void* d_out, int out_size, void* d_ws, size_t ws_size,
                              hipStream_t stream) {
    const float* x     = (const float*)d_in[0];
    const int*   edge  = (const int*)d_in[1];
    const int*   batch = (const int*)d_in[2];
    const float* W1    = (const float*)d_in[3];
    const float* b1    = (const float*)d_in[4];
    const float* W2    = (const float*)d_in[5];
    const float* b2    = (const float*)d_in[6];
    const float* gamma = (const float*)d_in[7];
    const float* beta  = (const float*)d_in[8];

    const int nNodes = in_sizes[0] / DIM;
    const int nEdges = in_sizes[1] / 2;

    float* pool   = (float*)d_out;                           // [NG, 3*DIM]
    float* hFinal = pool + (size_t)NG * NLAYERS * DIM;       // [nNodes, DIM]

    float* bufZ  = (float*)d_ws;                             // z / z2 (in-place MLP)
    float* bufH  = bufZ + (size_t)nNodes * DIM;              // current h
    float* accS  = bufH + (size_t)nNodes * DIM;              // [DIM] sum
    float* accQ  = accS + DIM;                               // [DIM] sumsq
    float* scale = accQ + DIM;                               // [DIM]
    float* shift = scale + DIM;                              // [DIM]

    k_zero<<<(NG * NLAYERS * DIM + 255) / 256, 256, 0, stream>>>(pool, NG * NLAYERS * DIM);

    const int n4       = nNodes * DIM / 4;
    const int tScatter = nEdges * 32;
    const int tNode    = nNodes * 32;

    for (int L = 0; L < NLAYERS; ++L) {
        const float* hIn = (L == 0) ? x : bufH;
        // z = h  (eps = 0 => (1+eps)*h)
        k_copy4<<<(n4 + 255) / 256, 256, 0, stream>>>((const float4*)hIn, (float4*)bufZ, n4);
        // z += segment_sum(h[src], dst)
        k_scatter<<<(tScatter + 255) / 256, 256, 0, stream>>>(hIn, edge, bufZ, nEdges);
        // zero BN accumulators (accS,accQ contiguous)
        k_zero<<<1, 256, 0, stream>>>(accS, 2 * DIM);
        // z = relu(relu(z@W1+b1)@W2+b2), in place, + BN statistics
        k_gin_mlp<<<(nNodes + 127) / 128, 256, LDS_BYTES, stream>>>(
            bufZ, W1 + (size_t)L * DIM * DIM, b1 + L * DIM,
            W2 + (size_t)L * DIM * DIM, b2 + L * DIM,
            bufZ, accS, accQ, nNodes);
        k_bn_finalize<<<1, DIM, 0, stream>>>(accS, accQ, gamma + L * DIM, beta + L * DIM,
                                             scale, shift, nNodes);
        float* hDst = (L == NLAYERS - 1) ? hFinal : bufH;
        k_bn_pool<<<(tNode + 255) / 256, 256, 0, stream>>>(bufZ, scale, shift, batch,
                                                           hDst, pool, L, nNodes);
    }
}